// NVARReservoir_48636209660445
// MI455X (gfx1250) — compile-verified
//
#include <hip/hip_runtime.h>
#include <hip/hip_bf16.h>

// NVAR quadratic-reservoir readout for MI455X (gfx1250, wave32).
//
// GEMM: [1024 tokens x 51681 feats] x [51681 x 64]  (fp32, ~6.8 GFLOP).
// Feature matrix generated on the fly from a 320-float/token LDS buffer:
// the quad block for fixed i is feats[m,i] * feats[m, i..319] (scalar x slice).
// fp32 WMMA V_WMMA_F32_16X16X4_F32; M-tile = 64 tokens (4 row-subtiles per
// wave, B-fragment reuse x4); K split 16-way across blockIdx.y with fp32
// global atomics combining partials in workspace h_pre.

#define NVAR_DELAY   5
#define NVAR_EMBED   64
#define NVAR_L       320            // DELAY*EMBED linear terms
#define NVAR_TOKENS  1024           // B*S = 2*512
#define NVAR_F       51681          // 1 + L + L*(L+1)/2
#define FSTRIDE      326            // LDS feats row stride (even, bank-spread)
#define KSLICES     16              // K-dimension split across blockIdx.y
#define MTOK        64              // tokens per workgroup
#define SUBT         4              // 16-row WMMA subtiles per wave (MTOK/16)

typedef __attribute__((ext_vector_type(2))) float v2f;
typedef __attribute__((ext_vector_type(8))) float v8f;

__global__ __launch_bounds__(256) void nvar_zero_kernel(float* __restrict__ hpre) {
    int idx = blockIdx.x * blockDim.x + threadIdx.x;
    if (idx < NVAR_TOKENS * NVAR_EMBED) hpre[idx] = 0.0f;
}

__global__ __launch_bounds__(128) void nvar_gemm1_kernel(const float* __restrict__ x,
                                                         const float* __restrict__ W1,
                                                         float* __restrict__ hpre) {
    __shared__ float feats[MTOK * FSTRIDE];   // 64 tokens x 320 feats (83.4 KB)

    const int t0    = blockIdx.x * MTOK;      // first token of this M-tile
    const int slice = blockIdx.y;             // K-slice (i % KSLICES == slice)

    // ---- Stage 1: delay-embedded linear features into LDS -----------------
    // feats[m][e*DELAY + k] = x[b, s + k - (DELAY-1), e]  (0 if out of range)
    for (int idx = threadIdx.x; idx < MTOK * NVAR_L; idx += 128) {
        int m = idx / NVAR_L;
        int f = idx - m * NVAR_L;
        int e = f / NVAR_DELAY;
        int k = f - e * NVAR_DELAY;
        int t = t0 + m;
        int b = t >> 9;                       // S = 512
        int s = t & 511;
        int sp = s + k - (NVAR_DELAY - 1);
        float v = (sp >= 0) ? x[((b << 9) + sp) * NVAR_EMBED + e] : 0.0f;
        feats[m * FSTRIDE + f] = v;
    }
    __syncthreads();

    // ---- Stage 2: WMMA K-loop --------------------------------------------
    const int lane = threadIdx.x & 31;
    const int wave = threadIdx.x >> 5;        // 4 waves = 4 column groups (N=64)
    const int c0   = wave * 16;
    const int mrow = lane & 15;               // A row within subtile / B column
    const int kofs = (lane >> 4) * 2;         // this half-wave holds K {kofs, kofs+1}
    const float* __restrict__ W1c = W1 + c0 + mrow;   // column-shifted W1 base

    v8f acc[SUBT];
    const float* fr[SUBT];
    #pragma unroll
    for (int s = 0; s < SUBT; ++s) {
        acc[s] = (v8f){0.f, 0.f, 0.f, 0.f, 0.f, 0.f, 0.f, 0.f};
        fr[s]  = feats + (s * 16 + mrow) * FSTRIDE;
    }

    if (slice == 0) {
        // linear terms: feature rows 1..320, K-step 4 (80 clamp-free steps)
        for (int p = 0; p < NVAR_L; p += 4) {
            int k0 = p + kofs;
            v2f bfr;
            bfr.x = W1c[(1 + k0) * NVAR_EMBED];
            bfr.y = W1c[(2 + k0) * NVAR_EMBED];
            #pragma unroll
            for (int s = 0; s < SUBT; ++s) {
                const float2 f = *(const float2*)(fr[s] + k0);  // 8B aligned
                v2f a; a.x = f.x; a.y = f.y;
                acc[s] = __builtin_amdgcn_wmma_f32_16x16x4_f32(false, a, false, bfr,
                                                               (short)0, acc[s], false, false);
            }
        }
    }

    // quadratic terms: segment i holds features feats[i]*feats[i..319];
    // flat W1 row base qb = 321 + 320*i - i*(i-1)/2.  i interleaved by slice.
    for (int i = slice; i < NVAR_L; i += KSLICES) {
        const int len = NVAR_L - i;
        const int qb  = 1 + NVAR_L + i * NVAR_L - (i * (i - 1)) / 2;
        float sv[SUBT];
        #pragma unroll
        for (int s = 0; s < SUBT; ++s) sv[s] = fr[s][i];

        const float* __restrict__ Bp = W1c + (qb + kofs) * NVAR_EMBED;
        const int len4 = len & ~3;            // clamp-free portion
        int p = 0;
        for (; p < len4; p += 4) {            // steady state: no clamps/selects
            v2f bfr;
            bfr.x = Bp[p * NVAR_EMBED];
            bfr.y = Bp[p * NVAR_EMBED + NVAR_EMBED];
            const int j = i + p + kofs;
            #pragma unroll
            for (int s = 0; s < SUBT; ++s) {
                v2f a;
                a.x = fr[s][j]     * sv[s];
                a.y = fr[s][j + 1] * sv[s];
                acc[s] = __builtin_amdgcn_wmma_f32_16x16x4_f32(false, a, false, bfr,
                                                               (short)0, acc[s], false, false);
            }
        }
        if (p < len) {                        // ragged tail (len % 4 in {1,2,3})
            int e0 = p + kofs, e1 = e0 + 1;
            int ce0 = (e0 < len) ? e0 : (len - 1);   // pad K positions get A=0,
            int ce1 = (e1 < len) ? e1 : (len - 1);   // so clamped B row is moot
            v2f bfr;
            bfr.x = W1c[(qb + ce0) * NVAR_EMBED];
            bfr.y = W1c[(qb + ce1) * NVAR_EMBED];
            #pragma unroll
            for (int s = 0; s < SUBT; ++s) {
                v2f a;
                a.x = (e0 < len) ? fr[s][i + ce0] * sv[s] : 0.0f;
                a.y = (e1 < len) ? fr[s][i + ce1] * sv[s] : 0.0f;
                acc[s] = __builtin_amdgcn_wmma_f32_16x16x4_f32(false, a, false, bfr,
                                                               (short)0, acc[s], false, false);
            }
        }
    }

    // ---- Stage 3: combine K-slice partials (fp32 global atomics) ----------
    // C/D tile layout: lanes 0-15 -> rows 0-7 in VGPR 0-7, lanes 16-31 -> rows 8-15.
    const int rbase = (lane >> 4) * 8;
    #pragma unroll
    for (int s = 0; s < SUBT; ++s) {
        #pragma unroll
        for (int v = 0; v < 8; ++v) {
            atomicAdd(&hpre[(t0 + s * 16 + rbase + v) * NVAR_EMBED + c0 + mrow], acc[s][v]);
        }
    }
}

__global__ __launch_bounds__(256) void nvar_readout_kernel(const float* __restrict__ hpre,
                                                           const float* __restrict__ W1,
                                                           const float* __restrict__ b1,
                                                           const float* __restrict__ W2,
                                                           const float* __restrict__ b2,
                                                           float* __restrict__ out) {
    __shared__ float hsh[64 * 65];
    const int t0 = blockIdx.x * 64;

    // const-feature row (W1[0,:]) + bias + exact GELU
    for (int idx = threadIdx.x; idx < 64 * NVAR_EMBED; idx += 256) {
        int r = idx >> 6, c = idx & 63;
        float hp = hpre[(t0 + r) * NVAR_EMBED + c] + W1[c] + b1[c];
        float h  = 0.5f * hp * (1.0f + erff(hp * 0.70710678118654752f));
        hsh[r * 65 + c] = h;
    }
    __syncthreads();

    // tiny 64x64 readout
    for (int idx = threadIdx.x; idx < 64 * NVAR_EMBED; idx += 256) {
        int r = idx >> 6, c = idx & 63;
        float o = b2[c];
        #pragma unroll
        for (int d = 0; d < NVAR_EMBED; ++d)
            o += hsh[r * 65 + d] * W2[d * NVAR_EMBED + c];
        out[(t0 + r) * NVAR_EMBED + c] = o;
    }
}

extern "C" void kernel_launch(void* const* d_in, const int* in_sizes, int n_in,
                              void* d_out, int out_size, void* d_ws, size_t ws_size,
                              hipStream_t stream) {
    const float* x  = (const float*)d_in[0];   // [2,512,64]
    const float* W1 = (const float*)d_in[1];   // [51681,64]
    const float* b1 = (const float*)d_in[2];   // [64]
    const float* W2 = (const float*)d_in[3];   // [64,64]
    const float* b2 = (const float*)d_in[4];   // [64]
    float* out  = (float*)d_out;               // [2,512,64]
    float* hpre = (float*)d_ws;                // 1024*64 fp32 partial pre-activations

    nvar_zero_kernel<<<(NVAR_TOKENS * NVAR_EMBED + 255) / 256, 256, 0, stream>>>(hpre);

    dim3 grid(NVAR_TOKENS / MTOK, KSLICES);    // 16 M-tiles x 16 K-slices = 256 WGs
    nvar_gemm1_kernel<<<grid, 128, 0, stream>>>(x, W1, hpre);

    nvar_readout_kernel<<<NVAR_TOKENS / 64, 256, 0, stream>>>(hpre, W1, b1, W2, b2, out);
}